// crf_layer_71691594104894
// MI455X (gfx1250) — compile-verified
//
#include <hip/hip_runtime.h>

#define N_SAMP 512
#define T_LEN  512
#define C_CLS  64

#define LOG2E 1.44269504088896340736f
#define LN2   0.69314718055994530942f

typedef __attribute__((ext_vector_type(2))) float v2f;
typedef __attribute__((ext_vector_type(8))) float v8f;

// ---------------------------------------------------------------------------
// Kernel 1: gold score per sample (embarrassingly parallel gather-scan).
// gold[n] = logits[n,0,tags0] + sum_j T[tags0,j]
//         + sum_{i=0}^{L-1} ( T[tags[i+1],tags[i]] + logits[n,i+1,tags[i+1]] )
// ---------------------------------------------------------------------------
__global__ __launch_bounds__(128) void crf_gold_kernel(
    const float* __restrict__ logits, const int* __restrict__ lengths,
    const int* __restrict__ tags, const float* __restrict__ Tm,
    float* __restrict__ gold)
{
    int n = blockIdx.x * blockDim.x + threadIdx.x;
    if (n >= N_SAMP) return;
    const int*   tg = tags + n * T_LEN;
    const float* lg = logits + (size_t)n * T_LEN * C_CLS;
    int L  = lengths[n];
    int ta = tg[0];
    float s = lg[ta];
    const float* trow = Tm + ta * C_CLS;
    float rs = 0.0f;
    #pragma unroll 8
    for (int j = 0; j < C_CLS; ++j) rs += trow[j];
    s += rs;
    for (int i = 0; i < L; ++i) {
        int tb = tg[i + 1];
        s += Tm[tb * C_CLS + ta] + lg[(size_t)(i + 1) * C_CLS + tb];
        ta = tb;
    }
    gold[n] = s;
}

// ---------------------------------------------------------------------------
// Kernel 2: CRF forward (partition) via WMMA.
// Per workgroup: 16 samples, 128 threads = 4 waves.  Per time step:
//   S(64x16) = exp(T)(64x64) x V(64x16),  V = exp(alpha - max)
// as 4 M-tiles x (2 independent 8-deep chains) of V_WMMA_F32_16X16X4_F32.
// alpha_new[i] = logit/64 + m_old + log(S[i]);  partition = sum_i alpha at
// t == len-1;  out[n] = partition - gold[n].
// Each thread owns sample column ln = lane&15 and 8 class rows, so the
// per-sample running max is a plain register (m_carry); the cross-class
// reduction is 8 register ops + one 8-partial LDS combine.  2 barriers/step.
// ---------------------------------------------------------------------------
__global__ __launch_bounds__(128) void crf_partition_kernel(
    const float* __restrict__ logits, const int* __restrict__ lengths,
    const float* __restrict__ Tm, const float* __restrict__ gold,
    float* __restrict__ out)
{
    __shared__ float Et[C_CLS * C_CLS];   // exp(transition), 16 KB (init only)
    __shared__ float Vbuf[C_CLS * 16];    // exp(alpha - m), [class][sample]
    __shared__ float Pmax[8 * 16];        // per-(wave,half) partial max
    __shared__ float Psum[8 * 16];        // per-(wave,half) partial sum
    __shared__ int   maxlen_s;

    const int tid  = threadIdx.x;
    const int wave = tid >> 5;
    const int lane = tid & 31;
    const int hi   = lane >> 4;   // half-wave select
    const int ln   = lane & 15;   // sample column
    const int n0   = blockIdx.x * 16;

    for (int e = tid; e < C_CLS * C_CLS; e += 128)
        Et[e] = __builtin_amdgcn_exp2f(Tm[e] * LOG2E);
    for (int e = tid; e < C_CLS * 16; e += 128) Vbuf[e] = 1.0f;  // exp(0-0)
    if (tid == 0) {
        int ml = 0;
        for (int i = 0; i < 16; ++i) ml = max(ml, lengths[n0 + i]);
        maxlen_s = ml;
    }
    __syncthreads();

    // Preload loop-invariant A fragments: Et rows [16*wave .. 16*wave+15].
    // ISA A 16x4 layout: lanes 0-15 -> K={0,1}, lanes 16-31 -> K={2,3}.
    v2f Afrag[16];
    {
        const int row = wave * 16 + ln;
        #pragma unroll
        for (int k = 0; k < 16; ++k) {
            const int kb = 4 * k + 2 * hi;
            v2f a;
            a.x = Et[row * C_CLS + kb];
            a.y = Et[row * C_CLS + kb + 1];
            Afrag[k] = a;
        }
    }

    const int maxlen = maxlen_s;
    int mylen = 0;
    if (tid < 16) mylen = lengths[n0 + tid];
    float part    = 0.0f;
    float m_carry = 0.0f;     // running max of alpha for sample ln

    // Each lane covers sample ln, classes [16*wave + 8*hi .. +7] of D tile.
    const int    rbase = wave * 16 + hi * 8;
    const size_t samp_base = ((size_t)(n0 + ln) * T_LEN) * C_CLS + rbase;
    const float  inv = 1.0f / (float)C_CLS;
    const int    pidx = (wave * 2 + hi) * 16 + ln;

    for (int t = 0; t < maxlen; ++t) {
        // ---- Phase A: S = Et*V via 2 independent WMMA chains ----
        v8f d0 = {0.f, 0.f, 0.f, 0.f, 0.f, 0.f, 0.f, 0.f};
        v8f d1 = {0.f, 0.f, 0.f, 0.f, 0.f, 0.f, 0.f, 0.f};
        #pragma unroll
        for (int k = 0; k < 8; ++k) {
            const int kb0 = 4 * (2 * k)     + 2 * hi;
            const int kb1 = 4 * (2 * k + 1) + 2 * hi;
            v2f b0, b1;
            b0.x = Vbuf[kb0 * 16 + ln];
            b0.y = Vbuf[(kb0 + 1) * 16 + ln];
            b1.x = Vbuf[kb1 * 16 + ln];
            b1.y = Vbuf[(kb1 + 1) * 16 + ln];
            d0 = __builtin_amdgcn_wmma_f32_16x16x4_f32(
                     false, Afrag[2 * k],     false, b0, (short)0, d0, false, false);
            d1 = __builtin_amdgcn_wmma_f32_16x16x4_f32(
                     false, Afrag[2 * k + 1], false, b1, (short)0, d1, false, false);
        }

        const float* lgp = logits + samp_base + (size_t)t * C_CLS;
        float4 l0 = *(const float4*)(lgp);
        float4 l1 = *(const float4*)(lgp + 4);
        __builtin_prefetch(lgp + C_CLS, 0, 3);   // next time step's row

        // alpha_new = logit/64 + m_old + ln(S);  S > 0 always (max col -> 1.0)
        float al[8];
        #pragma unroll
        for (int r = 0; r < 8; ++r) {
            float lr = (r < 4) ? ((const float*)&l0)[r] : ((const float*)&l1)[r - 4];
            float sr = d0[r] + d1[r];
            al[r] = lr * inv + m_carry + __builtin_amdgcn_logf(sr) * LN2;
        }

        // local tree reduction of this thread's 8 classes
        float mx01 = fmaxf(al[0], al[1]), mx23 = fmaxf(al[2], al[3]);
        float mx45 = fmaxf(al[4], al[5]), mx67 = fmaxf(al[6], al[7]);
        float lmax = fmaxf(fmaxf(mx01, mx23), fmaxf(mx45, mx67));
        float lsum = ((al[0] + al[1]) + (al[2] + al[3])) +
                     ((al[4] + al[5]) + (al[6] + al[7]));
        Pmax[pidx] = lmax;
        Psum[pidx] = lsum;
        __syncthreads();

        // ---- Phase B: every thread combines the 8 partials of its sample ----
        float mnew = -3.402823466e+38f;
        float snew = 0.0f;
        #pragma unroll
        for (int p = 0; p < 8; ++p) {
            mnew = fmaxf(mnew, Pmax[p * 16 + ln]);
            snew += Psum[p * 16 + ln];
        }
        // V = exp(alpha - m_new) from in-register alpha
        #pragma unroll
        for (int r = 0; r < 8; ++r)
            Vbuf[(rbase + r) * 16 + ln] =
                __builtin_amdgcn_exp2f((al[r] - mnew) * LOG2E);
        if (tid < 16 && t == mylen - 1) part = snew;  // partition[n]
        m_carry = mnew;
        __syncthreads();
    }

    if (tid < 16) out[n0 + tid] = part - gold[n0 + tid];
}

// ---------------------------------------------------------------------------
extern "C" void kernel_launch(void* const* d_in, const int* in_sizes, int n_in,
                              void* d_out, int out_size, void* d_ws, size_t ws_size,
                              hipStream_t stream)
{
    const float* logits  = (const float*)d_in[0];   // [N,T,C] f32
    const int*   lengths = (const int*)d_in[1];     // [N] i32
    const int*   tags    = (const int*)d_in[2];     // [N,T] i32
    const float* Tm      = (const float*)d_in[3];   // [C,C] f32
    float* out  = (float*)d_out;                    // [N] f32
    float* gold = (float*)d_ws;                     // [N] f32 scratch

    crf_gold_kernel<<<(N_SAMP + 127) / 128, 128, 0, stream>>>(
        logits, lengths, tags, Tm, gold);
    crf_partition_kernel<<<N_SAMP / 16, 128, 0, stream>>>(
        logits, lengths, Tm, gold, out);
}